// MemoryAccess_39118562132415
// MI455X (gfx1250) — compile-verified
//
#include <hip/hip_runtime.h>
#include <cmath>

// ---------------- problem constants ----------------
#define BB   2048
#define DD   256
#define NN   128
#define WSZ  64
#define RR   4
#define PRE  480            // padded 471 -> 480 (= 30 column tiles of 16)
#define EPSV 1e-6f

// pre-activation buffer column offsets (packed param heads)
#define OW_WV   0           // write_vectors   64
#define OW_EV   64          // erase_vectors   64
#define OW_FG   128         // free_gate        4
#define OW_AG   132         // alloc_gate       1
#define OW_WG   133         // write_gate       1
#define OW_RM   134         // read_mode       12
#define OW_WK   146         // write_keys      64
#define OW_WSTR 210         // write_strengths  1
#define OW_RK   211         // read_keys      256
#define OW_RSTR 467         // read_strengths   4

// output offsets (flat concat in reference return order)
#define O_WORDS 0ull
#define O_MEM   524288ull
#define O_RWN   17301504ull
#define O_WWN   18350080ull
#define O_LINK  18612224ull
#define O_PREC  52166656ull
#define O_USAGE 52428800ull

typedef float v2f __attribute__((ext_vector_type(2)));
typedef float v8f __attribute__((ext_vector_type(8)));

// D = A(16x4 f32) * B(4x16 f32) + C : exact f32 WMMA on the matrix pipe.
__device__ __forceinline__ v8f wmma4(v2f a, v2f b, v8f c) {
  return __builtin_amdgcn_wmma_f32_16x16x4_f32(false, a, false, b, (short)0, c,
                                               false, false);
}

__device__ __forceinline__ float sigmoidf_(float x) { return 1.0f / (1.0f + expf(-x)); }
__device__ __forceinline__ float softplusf_(float x) {
  return fmaxf(x, 0.0f) + log1pf(expf(-fabsf(x)));
}

// ---------------------------------------------------------------------------
// Pack helpers: W_pack[256 x 480], b_pack[480] (zero-padded, guard-free GEMM)
// ---------------------------------------------------------------------------
__global__ void dnc_zero(float* __restrict__ Wp, float* __restrict__ bp) {
  const int t = blockIdx.x * blockDim.x + threadIdx.x;
  if (t < DD * PRE) Wp[t] = 0.0f;
  if (t < PRE) bp[t] = 0.0f;
}

__global__ void dnc_pack(const float* __restrict__ W,
                         const float* __restrict__ bias,
                         float* __restrict__ Wp, float* __restrict__ bp,
                         int dim, int off) {
  const int t = blockIdx.x * blockDim.x + threadIdx.x;
  if (t < DD * dim) {
    const int k = t / dim, c = t - k * dim;
    Wp[(size_t)k * PRE + off + c] = W[t];
  }
  if (t < dim) bp[off + t] = bias[t];
}

// ---------------------------------------------------------------------------
// Kernel 1: pre = x @ W_pack + b_pack via f32 WMMA 16x16x4.
// 128 blocks (one 16-row x-tile in LDS), 8 waves x 4 column tiles each.
// Branch-free inner loop: ds_load_b64 (A) + 2x global_load_b32 (B) + wmma.
// ---------------------------------------------------------------------------
__global__ __launch_bounds__(256) void dnc_linear_wmma(
    const float* __restrict__ X, const float* __restrict__ Wp,
    const float* __restrict__ bp, float* __restrict__ pre) {
  __shared__ float s_x[16 * DD];  // 16 KB

  const int tid  = threadIdx.x;
  const int lane = tid & 31;
  const int wave = tid >> 5;
  const int half = lane >> 4;
  const int lm   = lane & 15;
  const int m0   = blockIdx.x * 16;

  {  // stage the 16x256 x tile with b128 transactions
    const float4* xsrc = reinterpret_cast<const float4*>(X + (size_t)m0 * DD);
    float4* xdst = reinterpret_cast<float4*>(s_x);
#pragma unroll
    for (int t = 0; t < (16 * DD / 4) / 256; ++t) xdst[tid + t * 256] = xsrc[tid + t * 256];
  }
  __syncthreads();

#pragma unroll
  for (int t = 0; t < 4; ++t) {
    const int tile = wave * 4 + t;         // 0..31; 30,31 skipped (wave-uniform)
    if (tile >= PRE / 16) break;
    const int col = tile * 16 + lm;
    v8f acc = {};
#pragma unroll 4
    for (int k = 0; k < DD; k += 4) {
      const int ka = k + 2 * half;
      v2f a, bmat;
      a.x = s_x[lm * DD + ka];
      a.y = s_x[lm * DD + ka + 1];
      bmat.x = Wp[(size_t)ka * PRE + col];
      bmat.y = Wp[(size_t)(ka + 1) * PRE + col];
      acc = wmma4(a, bmat, acc);
    }
    const float bv = bp[col];
#pragma unroll
    for (int g = 0; g < 8; ++g)
      pre[(size_t)(m0 + g + 8 * half) * PRE + tile * 16 + lm] = acc[g] + bv;
  }
}

// ---------------------------------------------------------------------------
// Kernel 2: fully fused per-batch DNC step. 1 workgroup (256 thr) per b.
// ---------------------------------------------------------------------------
__global__ __launch_bounds__(256) void dnc_fused(
    const float* __restrict__ mem_in, const float* __restrict__ rw_in,
    const float* __restrict__ ww_in, const float* __restrict__ link_in,
    const float* __restrict__ pw_in, const float* __restrict__ usage_in,
    const float* __restrict__ pre, float* __restrict__ out) {
  const int b    = blockIdx.x;
  const int tid  = threadIdx.x;
  const int lane = tid & 31;
  const int wave = tid >> 5;
  const int half = lane >> 4;
  const int lm   = lane & 15;

  __shared__ float s_pre[PRE];
  __shared__ float s_wv[WSZ], s_ev[WSZ], s_wkey[WSZ];
  __shared__ float s_rkey[RR * WSZ];
  __shared__ float s_fg[RR], s_rstr[RR], s_rknorm[RR];
  __shared__ float s_rmode[RR * 3];
  __shared__ float s_rw[RR * NN];       // prev read weights (WMMA A source)
  __shared__ float s_pw[NN];            // prev precedence
  __shared__ float s_nonusage[NN];
  __shared__ float s_wcw[NN];           // write content: sim -> softmax weights
  __shared__ float s_sorted[NN], s_cum[NN], s_alloc[NN];
  __shared__ float s_wwnew[NN];
  __shared__ float s_dotP[NN * 2 * RR]; // read-key dot partials
  __shared__ float s_normP[NN * 2];
  __shared__ float s_csim[RR * NN];
  __shared__ float s_link[64 * NN];     // half of link_new (32 KB)
  __shared__ float s_fwd[RR * NN], s_bwd[RR * NN];
  __shared__ float s_rwn[RR * NN];
  __shared__ float s_words[RR * WSZ];
  __shared__ float s_red[128];
  __shared__ float s_sc[16];            // 0:ag 1:wg 2:wstr 3:|wkey| 4:max 5:invsum 6..9:rmax 10..13:rinv

  // ---- Phase 0: stage pre-activations + prev read weights + precedence ----
  for (int t = tid; t < PRE; t += 256) s_pre[t] = pre[(size_t)b * PRE + t];
  for (int t = tid; t < RR * NN; t += 256) s_rw[t] = rw_in[(size_t)b * RR * NN + t];
  for (int t = tid; t < NN; t += 256) s_pw[t] = pw_in[(size_t)b * NN + t];
  __syncthreads();

  if (tid < WSZ) {
    s_wv[tid]   = s_pre[OW_WV + tid];
    s_ev[tid]   = sigmoidf_(s_pre[OW_EV + tid]);
    s_wkey[tid] = s_pre[OW_WK + tid];
  }
  s_rkey[tid] = s_pre[OW_RK + tid];  // exactly 256 threads = R*WS
  if (tid < RR) {
    s_fg[tid]   = sigmoidf_(s_pre[OW_FG + tid]);
    s_rstr[tid] = softplusf_(s_pre[OW_RSTR + tid]);
    float a0 = s_pre[OW_RM + tid * 3 + 0];
    float a1 = s_pre[OW_RM + tid * 3 + 1];
    float a2 = s_pre[OW_RM + tid * 3 + 2];
    float mx = fmaxf(a0, fmaxf(a1, a2));
    float e0 = expf(a0 - mx), e1 = expf(a1 - mx), e2 = expf(a2 - mx);
    float inv = 1.0f / (e0 + e1 + e2);
    s_rmode[tid * 3 + 0] = e0 * inv;
    s_rmode[tid * 3 + 1] = e1 * inv;
    s_rmode[tid * 3 + 2] = e2 * inv;
    float s = 0.0f;
    for (int w = 0; w < WSZ; ++w) { float v = s_pre[OW_RK + tid * WSZ + w]; s += v * v; }
    s_rknorm[tid] = sqrtf(s);
  }
  if (tid == 0) {
    s_sc[0] = sigmoidf_(s_pre[OW_AG]);
    s_sc[1] = sigmoidf_(s_pre[OW_WG]);
    s_sc[2] = softplusf_(s_pre[OW_WSTR]);
    float s = 0.0f;
    for (int w = 0; w < WSZ; ++w) { float v = s_pre[OW_WK + w]; s += v * v; }
    s_sc[3] = sqrtf(s);
  }
  __syncthreads();

  // ---- Phase 1: usage update; Phase 2a: write-content dot/norm ----
  if (tid < NN) {
    const int n = tid;
    float us = usage_in[(size_t)b * NN + n];
    float ww = ww_in[(size_t)b * NN + n];          // NW==1: 1-prod(1-ww)=ww
    float u  = us + (1.0f - us) * ww;
    float phi = 1.0f;
#pragma unroll
    for (int r = 0; r < RR; ++r) phi *= (1.0f - s_fg[r] * s_rw[r * NN + n]);
    float un = u * phi;
    out[O_USAGE + (size_t)b * NN + n] = un;
    s_nonusage[n] = 1.0f - (EPSV + (1.0f - EPSV) * un);

    float dot = 0.0f, nn2 = 0.0f;
    const float4* mrow4 =
        reinterpret_cast<const float4*>(mem_in + (size_t)b * NN * WSZ + (size_t)n * WSZ);
#pragma unroll
    for (int q = 0; q < WSZ / 4; ++q) {
      const float4 m = mrow4[q];
      dot += s_wkey[4 * q + 0] * m.x + s_wkey[4 * q + 1] * m.y +
             s_wkey[4 * q + 2] * m.z + s_wkey[4 * q + 3] * m.w;
      nn2 += m.x * m.x + m.y * m.y + m.z * m.z + m.w * m.w;
    }
    s_wcw[n] = dot;
    s_cum[n] = sqrtf(nn2);   // temp: old-memory row norms
  }
  __syncthreads();
  if (tid < NN) s_wcw[tid] = (s_wcw[tid] / (s_sc[3] * s_cum[tid] + EPSV)) * s_sc[2];
  __syncthreads();
  if (tid == 0) {           // write-content softmax scalars (N=128, cheap)
    float mx = -1e30f;
    for (int n = 0; n < NN; ++n) mx = fmaxf(mx, s_wcw[n]);
    float sm = 0.0f;
    for (int n = 0; n < NN; ++n) sm += expf(s_wcw[n] - mx);
    s_sc[4] = mx; s_sc[5] = 1.0f / sm;
  }
  __syncthreads();

  // ---- Phase 3: allocation (rank sort + exclusive cumprod) ----
  int rank = 0;
  if (tid < NN) {
    s_wcw[tid] = expf(s_wcw[tid] - s_sc[4]) * s_sc[5];   // finalize softmax
    const float v = s_nonusage[tid];
    for (int j = 0; j < NN; ++j) {
      const float vj = s_nonusage[j];
      rank += (vj > v) || (vj == v && j < tid);
    }
    s_sorted[rank] = v;
  }
  __syncthreads();
  if (tid == 0) {
    float run = 1.0f;
    for (int k = 0; k < NN; ++k) { s_cum[k] = s_sorted[k] * run; run *= (1.0f - s_sorted[k]); }
  }
  __syncthreads();
  if (tid < NN) s_alloc[tid] = s_cum[rank];
  __syncthreads();

  // ---- Phase 4: write weights, precedence ----
  if (tid < NN) {
    const float ag = s_sc[0], wg = s_sc[1];
    const float w = wg * (ag * s_alloc[tid] + (1.0f - ag) * s_wcw[tid]);
    s_wwnew[tid] = w;
    out[O_WWN + (size_t)b * NN + tid] = w;
  }
  __syncthreads();
  if (tid < 128) s_red[tid] = s_wwnew[tid];
  __syncthreads();
  for (int s = 64; s > 0; s >>= 1) { if (tid < s) s_red[tid] += s_red[tid + s]; __syncthreads(); }
  const float wsum = s_red[0];
  if (tid < NN) out[O_PREC + (size_t)b * NN + tid] = (1.0f - wsum) * s_pw[tid] + s_wwnew[tid];
  __syncthreads();

  // ---- Phase 5: memory update (rows held in registers) + read-key dots ----
  float mreg[32];
  {
    const int n = tid >> 1, h = tid & 1;
    const float4* mrow4 = reinterpret_cast<const float4*>(
        mem_in + (size_t)b * NN * WSZ + (size_t)n * WSZ + h * 32);
    float4* orow4 = reinterpret_cast<float4*>(
        out + O_MEM + (size_t)b * NN * WSZ + (size_t)n * WSZ + h * 32);
    const float wwn = s_wwnew[n];
    float dp0 = 0, dp1 = 0, dp2 = 0, dp3 = 0, nr = 0;
#pragma unroll
    for (int q = 0; q < 8; ++q) {
      const float4 m = mrow4[q];
      float4 nv;
#pragma unroll
      for (int c = 0; c < 4; ++c) {
        const int w = h * 32 + q * 4 + c;
        const float mv = (c == 0) ? m.x : (c == 1) ? m.y : (c == 2) ? m.z : m.w;
        const float v = mv * (1.0f - wwn * s_ev[w]) + wwn * s_wv[w];
        mreg[q * 4 + c] = v;
        if (c == 0) nv.x = v; else if (c == 1) nv.y = v; else if (c == 2) nv.z = v; else nv.w = v;
        dp0 += s_rkey[0 * WSZ + w] * v;
        dp1 += s_rkey[1 * WSZ + w] * v;
        dp2 += s_rkey[2 * WSZ + w] * v;
        dp3 += s_rkey[3 * WSZ + w] * v;
        nr  += v * v;
      }
      orow4[q] = nv;
    }
    const int p = (n * 2 + h) * RR;
    s_dotP[p + 0] = dp0; s_dotP[p + 1] = dp1; s_dotP[p + 2] = dp2; s_dotP[p + 3] = dp3;
    s_normP[n * 2 + h] = nr;
  }
  __syncthreads();
  if (tid < NN) {
    const int n = tid;
    const float nrm = sqrtf(s_normP[2 * n] + s_normP[2 * n + 1]);
#pragma unroll
    for (int r = 0; r < RR; ++r) {
      const float dot = s_dotP[(2 * n) * RR + r] + s_dotP[(2 * n + 1) * RR + r];
      s_csim[r * NN + n] = (dot / (s_rknorm[r] * nrm + EPSV)) * s_rstr[r];
    }
  }

  // ---- Phase 6: link update (two 64-row passes in LDS) + WMMA directional --
  v8f accB = {};  // bwd accumulator, tile c0 = wave*16, K accumulated over both passes
  for (int p = 0; p < 2; ++p) {
    __syncthreads();
    {
      const int i  = p * 64 + (tid >> 2);
      const int j0 = (tid & 3) * 32;
      const float wi  = s_wwnew[i];
      const float wip = wi;  // ww_i * pw_j factor uses same wi
      const float4* lrow4 = reinterpret_cast<const float4*>(
          link_in + (size_t)b * NN * NN + (size_t)i * NN + j0);
      float4* lout4 = reinterpret_cast<float4*>(
          out + O_LINK + (size_t)b * NN * NN + (size_t)i * NN + j0);
      float4* sl4 = reinterpret_cast<float4*>(&s_link[(i - p * 64) * NN + j0]);
#pragma unroll
      for (int q = 0; q < 8; ++q) {
        const float4 lv = lrow4[q];
        float4 nl;
#pragma unroll
        for (int c = 0; c < 4; ++c) {
          const int j = j0 + q * 4 + c;
          const float lvv = (c == 0) ? lv.x : (c == 1) ? lv.y : (c == 2) ? lv.z : lv.w;
          float v = (1.0f - wi - s_wwnew[j]) * lvv + wip * s_pw[j];
          if (i == j) v = 0.0f;
          if (c == 0) nl.x = v; else if (c == 1) nl.y = v; else if (c == 2) nl.z = v; else nl.w = v;
        }
        sl4[q]   = nl;
        lout4[q] = nl;
      }
    }
    __syncthreads();
    // bwd[r,j] += sum_{ii<64} rw[r, p*64+ii] * L[p*64+ii, j]   (all 8 waves)
    {
      const int c0 = wave * 16;
#pragma unroll 4
      for (int k = 0; k < 64; k += 4) {
        const int ka = k + 2 * half;
        v2f a, bmat;
        a.x = (lm < RR) ? s_rw[lm * NN + p * 64 + ka] : 0.0f;
        a.y = (lm < RR) ? s_rw[lm * NN + p * 64 + ka + 1] : 0.0f;
        bmat.x = s_link[ka * NN + c0 + lm];
        bmat.y = s_link[(ka + 1) * NN + c0 + lm];
        accB = wmma4(a, bmat, accB);
      }
    }
    // fwd[r,j] = sum_i rw[r,i] * L[j,i] for j in this pass's rows (waves 0..3)
    if (wave < 4) {
      v8f accF = {};
      const int c0loc = wave * 16;
#pragma unroll 4
      for (int k = 0; k < NN; k += 4) {
        const int ka = k + 2 * half;
        v2f a, bmat;
        a.x = (lm < RR) ? s_rw[lm * NN + ka] : 0.0f;
        a.y = (lm < RR) ? s_rw[lm * NN + ka + 1] : 0.0f;
        bmat.x = s_link[(c0loc + lm) * NN + ka];       // transposed LDS read
        bmat.y = s_link[(c0loc + lm) * NN + ka + 1];
        accF = wmma4(a, bmat, accF);
      }
#pragma unroll
      for (int g = 0; g < 8; ++g) {
        const int row = g + 8 * half;
        if (row < RR) s_fwd[row * NN + p * 64 + c0loc + lm] = accF[g];
      }
    }
  }
  {
    const int c0 = wave * 16;
#pragma unroll
    for (int g = 0; g < 8; ++g) {
      const int row = g + 8 * half;
      if (row < RR) s_bwd[row * NN + c0 + lm] = accB[g];
    }
  }
  __syncthreads();

  // ---- Phase 7: read content softmax + new read weights ----
  if (tid < RR) {
    float mx = -1e30f;
    for (int n = 0; n < NN; ++n) mx = fmaxf(mx, s_csim[tid * NN + n]);
    float sm = 0.0f;
    for (int n = 0; n < NN; ++n) sm += expf(s_csim[tid * NN + n] - mx);
    s_sc[6 + tid]  = mx;
    s_sc[10 + tid] = 1.0f / sm;
  }
  __syncthreads();
  if (tid < NN) {
#pragma unroll
    for (int r = 0; r < RR; ++r) {
      const float cw = expf(s_csim[r * NN + tid] - s_sc[6 + r]) * s_sc[10 + r];
      const float bm = s_rmode[r * 3 + 0], fm = s_rmode[r * 3 + 1], cm = s_rmode[r * 3 + 2];
      const float v = cm * cw + fm * s_fwd[r * NN + tid] + bm * s_bwd[r * NN + tid];
      s_rwn[r * NN + tid] = v;
      out[O_RWN + (size_t)b * RR * NN + r * NN + tid] = v;
    }
  }
  s_words[tid] = 0.0f;  // 256 == R*WS
  __syncthreads();

  // ---- Phase 8: read_words via LDS f32 atomics on register-held memory ----
  {
    const int n = tid >> 1, h = tid & 1;
#pragma unroll
    for (int r = 0; r < RR; ++r) {
      const float c = s_rwn[r * NN + n];
#pragma unroll
      for (int k = 0; k < 32; ++k)
        atomicAdd(&s_words[r * WSZ + h * 32 + k], c * mreg[k]);
    }
  }
  __syncthreads();
  out[O_WORDS + (size_t)b * RR * WSZ + tid] = s_words[tid];
}

// ---------------------------------------------------------------------------
extern "C" void kernel_launch(void* const* d_in, const int* in_sizes, int n_in,
                              void* d_out, int out_size, void* d_ws,
                              size_t ws_size, hipStream_t stream) {
  const float* x     = (const float*)d_in[0];
  const float* mem   = (const float*)d_in[1];
  const float* rw    = (const float*)d_in[2];
  const float* ww    = (const float*)d_in[3];
  const float* link  = (const float*)d_in[4];
  const float* pw    = (const float*)d_in[5];
  const float* usage = (const float*)d_in[6];
  float* out = (float*)d_out;

  // workspace layout: W_pack | b_pack | pre
  float* Wp  = (float*)d_ws;                 // 256*480 floats
  float* bp  = Wp + DD * PRE;                // 480 floats
  float* pre = bp + PRE;                     // 2048*480 floats

  dnc_zero<<<(DD * PRE + 255) / 256, 256, 0, stream>>>(Wp, bp);

  struct Head { int Wi, bi, off, dim; };
  const Head heads[10] = {
      {7, 8, OW_WV, 64},   {9, 10, OW_EV, 64},  {11, 12, OW_FG, 4},
      {13, 14, OW_AG, 1},  {15, 16, OW_WG, 1},  {17, 18, OW_RM, 12},
      {19, 20, OW_WK, 64}, {21, 22, OW_WSTR, 1},{23, 24, OW_RK, 256},
      {25, 26, OW_RSTR, 4}};

  for (int i = 0; i < 10; ++i) {
    const int n = DD * heads[i].dim;
    dnc_pack<<<(n + 255) / 256, 256, 0, stream>>>(
        (const float*)d_in[heads[i].Wi], (const float*)d_in[heads[i].bi], Wp,
        bp, heads[i].dim, heads[i].off);
  }

  dnc_linear_wmma<<<BB / 16, 256, 0, stream>>>(x, Wp, bp, pre);
  dnc_fused<<<BB, 256, 0, stream>>>(mem, rw, ww, link, pw, usage, pre, out);
}